// LogicLayer_15702400434247
// MI455X (gfx1250) — compile-verified
//
#include <hip/hip_runtime.h>
#include <cstdint>
#include <cstddef>

// Problem shape (fixed by the reference)
#define IN_DIM   256
#define OUT_DIM  256
#define B_DIM    4096
#define TB       16            // atom rows per block
#define IC       64            // i-chunk staged in LDS
#define NCHUNK   (IN_DIM / IC) // 4
#define NTHREADS 256
#define A_PITCH  16            // dwords per c-row of aT (== TB)

#if __has_builtin(__builtin_amdgcn_global_load_async_to_lds_b32)
#define HAVE_ASYNC 1
#else
#define HAVE_ASYNC 0
#endif

typedef __attribute__((address_space(1))) int GintT;  // global int
typedef __attribute__((address_space(3))) int LintT;  // LDS int

// ---------------------------------------------------------------------------
// Kernel 1: fw = sigmoid(weights) into workspace (pure-copy staging in the
// main kernel then qualifies for async global->LDS copies).
// ---------------------------------------------------------------------------
__global__ void sigmoid_precompute_kernel(const float* __restrict__ w,
                                          float* __restrict__ fw, int n) {
  int i = blockIdx.x * blockDim.x + threadIdx.x;
  if (i < n) fw[i] = 1.0f / (1.0f + expf(-w[i]));
}

// ---------------------------------------------------------------------------
// Kernel 2: y[b,o] = prod_i ( a[b,i]*fw[o,i] + (1 - fw[o,i]) )
// Block: TB atom rows x all OUT_DIM outputs.  Thread: 4x4 register tile.
// Async path: double-buffered LDS, copies of chunk ch+1 overlap compute of ch.
// ---------------------------------------------------------------------------
__global__ __launch_bounds__(NTHREADS)
void logic_prod_kernel(const float* __restrict__ atoms,
                       const float* __restrict__ wsrc,   // sigmoid'd or raw
                       int wsrc_is_sigmoid,
                       float* __restrict__ out) {
  __shared__ float fwT[2][IC * OUT_DIM];  // [buf][c][o]  2 x 64 KB
  __shared__ float aT[2][IC * A_PITCH];   // [buf][c][r]  2 x 4 KB

  const int tid = threadIdx.x;
  const int b0  = blockIdx.x * TB;
  const int tx  = tid & 63;   // 64 output tiles of 4
  const int ty  = tid >> 6;   // 4 row tiles of 4
  const int o0  = tx * 4;
  const int r0  = ty * 4;

  float p[16];
#pragma unroll
  for (int j = 0; j < 16; ++j) p[j] = 1.0f;

#if HAVE_ASYNC
  if (wsrc_is_sigmoid) {
    // ---------------- pipelined async path ----------------
    // Per-chunk async stage: transpose via per-lane LDS addresses, lanes walk
    // consecutive global addresses (coalesced requests into the async unit).
    auto issue_stage = [&](int ch, int buf) {
      const int i0 = ch * IC;
#pragma unroll 8
      for (int k = 0; k < (IC * OUT_DIM) / NTHREADS; ++k) {   // 64 per thread
        int flat = tid + k * NTHREADS;
        int c = flat & (IC - 1);
        int o = flat >> 6;
        const float* gp = wsrc + o * IN_DIM + i0 + c;
        float*       lp = &fwT[buf][c * OUT_DIM + o];
        __builtin_amdgcn_global_load_async_to_lds_b32(
            (GintT*)gp, (LintT*)lp, 0, 0);
      }
#pragma unroll
      for (int k = 0; k < (IC * TB) / NTHREADS; ++k) {        // 4 per thread
        int flat = tid + k * NTHREADS;
        int c = flat & (IC - 1);
        int r = flat >> 6;
        const float* gp = atoms + (size_t)(b0 + r) * IN_DIM + i0 + c;
        float*       lp = &aT[buf][c * A_PITCH + r];
        __builtin_amdgcn_global_load_async_to_lds_b32(
            (GintT*)gp, (LintT*)lp, 0, 0);
      }
    };

    issue_stage(0, 0);
    for (int ch = 0; ch < NCHUNK; ++ch) {
      const int buf = ch & 1;
      // Wait for this wave's outstanding copies (chunk ch), publish to block.
#if __has_builtin(__builtin_amdgcn_s_wait_asynccnt)
      __builtin_amdgcn_s_wait_asynccnt(0);
#else
      asm volatile("s_wait_asynccnt 0x0" ::: "memory");
#endif
      __syncthreads();
      // Overlap: start copying chunk ch+1 into the other buffer.
      if (ch + 1 < NCHUNK) issue_stage(ch + 1, buf ^ 1);

      // Compute chunk ch: 4x4 register tile, 1 FMA + 1 MUL per term.
#pragma unroll 4
      for (int c = 0; c < IC; ++c) {
        const float4 fw4 = *(const float4*)&fwT[buf][c * OUT_DIM + o0];
        const float4 a4  = *(const float4*)&aT[buf][c * A_PITCH + r0];
        float fv[4] = {fw4.x, fw4.y, fw4.z, fw4.w};
        float nv[4] = {1.0f - fw4.x, 1.0f - fw4.y, 1.0f - fw4.z, 1.0f - fw4.w};
        float av[4] = {a4.x, a4.y, a4.z, a4.w};
#pragma unroll
        for (int j = 0; j < 4; ++j)
#pragma unroll
          for (int k = 0; k < 4; ++k)
            p[j * 4 + k] *= fmaf(av[j], fv[k], nv[k]);   // a*fw + (1-fw)
      }
      // Next iteration's top-of-loop barrier protects buffer reuse.
    }
  } else
#endif
  {
    // ---------------- synchronous fallback (applies sigmoid while staging) --
    for (int ch = 0; ch < NCHUNK; ++ch) {
      const int i0 = ch * IC;
      // Lanes contiguous in o => conflict-free LDS writes; reads hit L2.
#pragma unroll 4
      for (int k = 0; k < (IC * OUT_DIM) / NTHREADS; ++k) {
        int flat = tid + k * NTHREADS;
        int o = flat & (OUT_DIM - 1);
        int c = flat >> 8;
        float v = wsrc[o * IN_DIM + i0 + c];
        if (!wsrc_is_sigmoid) v = 1.0f / (1.0f + expf(-v));
        fwT[0][c * OUT_DIM + o] = v;
      }
#pragma unroll
      for (int k = 0; k < (IC * TB) / NTHREADS; ++k) {
        int flat = tid + k * NTHREADS;
        int r = flat & (TB - 1);
        int c = flat >> 4;
        aT[0][c * A_PITCH + r] = atoms[(size_t)(b0 + r) * IN_DIM + i0 + c];
      }
      if (ch + 1 < NCHUNK) {  // global_prefetch_b8 of next chunk
        __builtin_prefetch(wsrc + (tid >> 6) * IN_DIM + i0 + IC + (tid & 63), 0, 1);
        __builtin_prefetch(atoms + (size_t)b0 * IN_DIM + i0 + IC + tid, 0, 1);
      }
      __syncthreads();
#pragma unroll 4
      for (int c = 0; c < IC; ++c) {
        const float4 fw4 = *(const float4*)&fwT[0][c * OUT_DIM + o0];
        const float4 a4  = *(const float4*)&aT[0][c * A_PITCH + r0];
        float fv[4] = {fw4.x, fw4.y, fw4.z, fw4.w};
        float nv[4] = {1.0f - fw4.x, 1.0f - fw4.y, 1.0f - fw4.z, 1.0f - fw4.w};
        float av[4] = {a4.x, a4.y, a4.z, a4.w};
#pragma unroll
        for (int j = 0; j < 4; ++j)
#pragma unroll
          for (int k = 0; k < 4; ++k)
            p[j * 4 + k] *= fmaf(av[j], fv[k], nv[k]);
      }
      __syncthreads();
    }
  }

  // ----- coalesced float4 output stores -----
#pragma unroll
  for (int j = 0; j < 4; ++j) {
    float4 v = make_float4(p[j * 4 + 0], p[j * 4 + 1], p[j * 4 + 2], p[j * 4 + 3]);
    *(float4*)&out[(size_t)(b0 + r0 + j) * OUT_DIM + o0] = v;
  }
}

// ---------------------------------------------------------------------------
extern "C" void kernel_launch(void* const* d_in, const int* in_sizes, int n_in,
                              void* d_out, int out_size, void* d_ws, size_t ws_size,
                              hipStream_t stream) {
  const float* atoms   = (const float*)d_in[0];   // (4096, 256)
  const float* weights = (const float*)d_in[1];   // (256, 256)
  float* out = (float*)d_out;                     // (4096, 256)

  const size_t fw_bytes = (size_t)OUT_DIM * IN_DIM * sizeof(float);
  const int use_ws = (d_ws != nullptr && ws_size >= fw_bytes) ? 1 : 0;

  const float* wsrc = weights;
  if (use_ws) {
    const int n = OUT_DIM * IN_DIM;
    sigmoid_precompute_kernel<<<(n + 255) / 256, 256, 0, stream>>>(
        weights, (float*)d_ws, n);
    wsrc = (const float*)d_ws;
  }

  logic_prod_kernel<<<B_DIM / TB, NTHREADS, 0, stream>>>(
      atoms, wsrc, use_ws, out);
}